// GIN_Custom_18777597018569
// MI455X (gfx1250) — compile-verified
//
#include <hip/hip_runtime.h>
#include <hip/hip_bf16.h>

typedef __attribute__((ext_vector_type(2))) float v2f;
typedef __attribute__((ext_vector_type(8))) float v8f;

#define DCH 128

// ---------------------------------------------------------------------------
// Scatter-add: agg[dst] += x[src] over all edges. One wave handles 32 edges.
// Lanes load 32 consecutive src/dst indices coalesced, broadcast each edge
// via __shfl; per edge each lane moves a contiguous float4 (wave = 512B row)
// and issues 4 global_atomic_add_f32.
// ---------------------------------------------------------------------------
__global__ __launch_bounds__(256) void scatter_add_kernel(
    const float* __restrict__ x,
    const long long* __restrict__ src,
    const long long* __restrict__ dst,
    float* __restrict__ agg,
    long long nE)
{
    const int lane = threadIdx.x & 31;
    const long long waveId = (long long)blockIdx.x * (blockDim.x >> 5) + (threadIdx.x >> 5);
    const long long e0 = waveId * 32;
    if (e0 >= nE) return;

    long long sIdx = 0, dIdx = 0;
    if (e0 + lane < nE) {
        sIdx = src[e0 + lane];
        dIdx = dst[e0 + lane];
    }
    const int cnt = (int)((nE - e0 < 32) ? (nE - e0) : 32);
    for (int i = 0; i < cnt; ++i) {
        const long long s = __shfl(sIdx, i, 32);
        const long long d = __shfl(dIdx, i, 32);
        const float4 v = *(const float4*)(x + s * DCH + (lane << 2));
        float* b = agg + d * DCH + (lane << 2);
        atomicAdd(b + 0, v.x);
        atomicAdd(b + 1, v.y);
        atomicAdd(b + 2, v.z);
        atomicAdd(b + 3, v.w);
    }
}

// ---------------------------------------------------------------------------
// Stage W:[128,128] row-major into LDS with K-pair interleaving:
// sW[(k>>1)*128 + n] = { W[k][n], W[k+1][n] }  -- exactly one WMMA B fragment,
// so the hot loop issues a single aligned ds_load_b64 per fragment.
// ---------------------------------------------------------------------------
__device__ __forceinline__ void stage_w_pairs(v2f* sW, const float* __restrict__ W, int tid)
{
    const float4* W4 = (const float4*)W;   // 4096 float4s: row r = f>>5, colquad = f&31
#pragma unroll
    for (int i = 0; i < 8; ++i) {
        const int q  = tid + 256 * i;      // 0..2047 : rowpair (q>>5), colquad (q&31)
        const int r2 = q >> 5;
        const int cq = q & 31;
        const float4 a = W4[(2 * r2) * 32 + cq];
        const float4 b = W4[(2 * r2 + 1) * 32 + cq];
        v2f* o = &sW[r2 * DCH + cq * 4];
        v2f p0; p0.x = a.x; p0.y = b.x;
        v2f p1; p1.x = a.y; p1.y = b.y;
        v2f p2; p2.x = a.z; p2.y = b.z;
        v2f p3; p3.x = a.w; p3.y = b.w;
        o[0] = p0; o[1] = p1; o[2] = p2; o[3] = p3;
    }
}

// ---------------------------------------------------------------------------
// Fused GEMM: Out = act(A @ W + bias [+ resid]), A:[N,128], W:[128,128] row-major.
// Block = 256 threads = 8 waves; block computes 128 rows x 128 cols.
// W staged in LDS (64KB, pair-interleaved). Each wave: 16 rows, 8 accumulator
// tiles (v8f), K-loop in steps of 4 using v_wmma_f32_16x16x4_f32 (exact fp32).
// MODE 0: ReLU.  MODE 1: + resid.  MODE 2: + resid then ELU.
// ---------------------------------------------------------------------------
template <int MODE>
__global__ __launch_bounds__(256) void gin_gemm_kernel(
    const float* __restrict__ A,
    const float* __restrict__ W,
    const float* __restrict__ bias,
    const float* __restrict__ resid,
    float* __restrict__ Out,
    int N)
{
    __shared__ v2f sW[64 * DCH];           // 64KB: K-pair-interleaved W
    const int tid = threadIdx.x;
    stage_w_pairs(sW, W, tid);
    __syncthreads();

    const int lane = tid & 31;
    const int wave = tid >> 5;
    const int half = lane >> 4;   // K-pair / M-half selector
    const int l16  = lane & 15;   // M (for A) or N (for B/C/D)
    const long long rowBase = (long long)blockIdx.x * 128 + wave * 16;
    if (rowBase + 16 > N) return;  // N % 16 == 0 -> wave-granular, after barrier

    v8f acc[8] = {};
    const float* arow = A + (rowBase + l16) * DCH;

    for (int k = 0; k < DCH; k += 4) {
        // A 16x4 frag: lane holds A[m][k+2*half], A[m][k+2*half+1]
        const v2f a = *(const v2f*)(arow + k + 2 * half);
        // B frags: pair row (k+2*half)/2 of interleaved LDS image
        const v2f* wr = &sW[((k + 2 * half) >> 1) * DCH + l16];
#pragma unroll
        for (int t = 0; t < 8; ++t) {
            const v2f b = wr[t * 16];      // { W[k+2h][n], W[k+2h+1][n] }
            acc[t] = __builtin_amdgcn_wmma_f32_16x16x4_f32(
                false, a, false, b, (short)0, acc[t], false, false);
        }
    }

    // Epilogue: D[m][n], m = rowBase + v + 8*half, n = t*16 + l16
#pragma unroll
    for (int t = 0; t < 8; ++t) {
        const int n = t * 16 + l16;
        const float bv = bias[n];
#pragma unroll
        for (int v = 0; v < 8; ++v) {
            const long long m = rowBase + v + 8 * half;
            float val = acc[t][v] + bv;
            if (MODE == 0) {
                val = fmaxf(val, 0.0f);
            } else {
                val += resid[m * DCH + n];
                if (MODE == 2) val = (val > 0.0f) ? val : expm1f(val);
            }
            Out[m * DCH + n] = val;
        }
    }
}

// ---------------------------------------------------------------------------
// JK projection: Out = [xs0|xs1|xs2] @ Wjk + bias, Wjk:[384,128] row-major.
// Same structure, looping over the 3 concat parts with W reloads in LDS.
// ---------------------------------------------------------------------------
__global__ __launch_bounds__(256) void jk_gemm_kernel(
    const float* __restrict__ x0,
    const float* __restrict__ x1,
    const float* __restrict__ x2,
    const float* __restrict__ Wjk,
    const float* __restrict__ bias,
    float* __restrict__ Out,
    int N)
{
    __shared__ v2f sW[64 * DCH];
    const int tid = threadIdx.x;
    const int lane = tid & 31;
    const int wave = tid >> 5;
    const int half = lane >> 4;
    const int l16  = lane & 15;
    const long long rowBase = (long long)blockIdx.x * 128 + wave * 16;
    const bool active = (rowBase + 16 <= N);

    v8f acc[8] = {};
    const float* parts[3] = {x0, x1, x2};

    for (int p = 0; p < 3; ++p) {
        __syncthreads();
        stage_w_pairs(sW, Wjk + (size_t)p * DCH * DCH, tid);
        __syncthreads();
        if (active) {
            const float* arow = parts[p] + (rowBase + l16) * DCH;
            for (int k = 0; k < DCH; k += 4) {
                const v2f a = *(const v2f*)(arow + k + 2 * half);
                const v2f* wr = &sW[((k + 2 * half) >> 1) * DCH + l16];
#pragma unroll
                for (int t = 0; t < 8; ++t) {
                    const v2f b = wr[t * 16];
                    acc[t] = __builtin_amdgcn_wmma_f32_16x16x4_f32(
                        false, a, false, b, (short)0, acc[t], false, false);
                }
            }
        }
    }
    if (!active) return;

#pragma unroll
    for (int t = 0; t < 8; ++t) {
        const int n = t * 16 + l16;
        const float bv = bias[n];
#pragma unroll
        for (int v = 0; v < 8; ++v) {
            const long long m = rowBase + v + 8 * half;
            Out[m * DCH + n] = acc[t][v] + bv;
        }
    }
}

// ---------------------------------------------------------------------------
extern "C" void kernel_launch(void* const* d_in, const int* in_sizes, int n_in,
                              void* d_out, int out_size, void* d_ws, size_t ws_size,
                              hipStream_t stream) {
    const float* x = (const float*)d_in[0];
    const long long* ei = (const long long*)d_in[1];
    const long long E = (long long)in_sizes[1] / 2;
    const int N = in_sizes[0] / DCH;
    const long long* srcI = ei;
    const long long* dstI = ei + E;

    const float* w1[3] = {(const float*)d_in[2], (const float*)d_in[6], (const float*)d_in[10]};
    const float* b1[3] = {(const float*)d_in[3], (const float*)d_in[7], (const float*)d_in[11]};
    const float* w2[3] = {(const float*)d_in[4], (const float*)d_in[8], (const float*)d_in[12]};
    const float* b2[3] = {(const float*)d_in[5], (const float*)d_in[9], (const float*)d_in[13]};
    const float* jkw = (const float*)d_in[14];
    const float* jkb = (const float*)d_in[15];

    const size_t bufElems = (size_t)N * DCH;
    float* agg = (float*)d_ws;
    float* h   = agg + bufElems;
    float* xs0 = h + bufElems;
    float* xs1 = xs0 + bufElems;
    float* xs2 = xs1 + bufElems;
    float* xs[3] = {xs0, xs1, xs2};

    const int gemmGrid = (N + 127) / 128;                 // 782 blocks
    const int scatGrid = (int)((E + 255) / 256);          // 256 edges per block (8 waves x 32)

    const float* xin = x;
    for (int l = 0; l < 3; ++l) {
        // agg = x  (the (1+eps)*x term, eps=0), then agg[dst] += x[src]
        hipMemcpyAsync(agg, xin, bufElems * sizeof(float), hipMemcpyDeviceToDevice, stream);
        scatter_add_kernel<<<scatGrid, 256, 0, stream>>>(xin, srcI, dstI, agg, E);
        // h = relu(agg @ w1 + b1)
        gin_gemm_kernel<0><<<gemmGrid, 256, 0, stream>>>(agg, w1[l], b1[l], nullptr, h, N);
        // xs[l] = h @ w2 + b2 + xin  (+ ELU except last layer)
        if (l < 2)
            gin_gemm_kernel<2><<<gemmGrid, 256, 0, stream>>>(h, w2[l], b2[l], xin, xs[l], N);
        else
            gin_gemm_kernel<1><<<gemmGrid, 256, 0, stream>>>(h, w2[l], b2[l], xin, xs[l], N);
        xin = xs[l];
    }
    // out = concat(xs) @ jk_w + jk_b
    jk_gemm_kernel<<<gemmGrid, 256, 0, stream>>>(xs0, xs1, xs2, jkw, jkb, (float*)d_out, N);
}